// NGCFLayer_55997783605424
// MI455X (gfx1250) — compile-verified
//
#include <hip/hip_runtime.h>

typedef float v2f __attribute__((ext_vector_type(2)));
typedef float v8f __attribute__((ext_vector_type(8)));

#define DIMD 256

// ---------------- zero the neighbor accumulator ----------------
__global__ __launch_bounds__(256) void zero_kernel(float4* __restrict__ p, int n4) {
    int i = blockIdx.x * 256 + threadIdx.x;
    if (i < n4) p[i] = make_float4(0.f, 0.f, 0.f, 0.f);
}

// ---------------- SpMM: one wave32 per edge --------------------
// nb[row] += val * emb[col]   (256 floats, hardware f32 atomics)
__global__ __launch_bounds__(256) void spmm_kernel(const float* __restrict__ emb,
                                                   const float* __restrict__ vals,
                                                   const int* __restrict__ rows,
                                                   const int* __restrict__ cols,
                                                   float* __restrict__ nb, int E) {
    const int e = blockIdx.x * 8 + (threadIdx.x >> 5);   // 8 waves per block
    if (e >= E) return;
    const int lane = threadIdx.x & 31;
    const int r = rows[e];
    const int c = cols[e];
    const float v = vals[e];
    const float4* src4 = (const float4*)(emb + (size_t)c * DIMD);
    float* dst = nb + (size_t)r * DIMD + lane * 8;
    float4 p0 = src4[lane * 2];       // two b128 loads cover the 1KB row
    float4 p1 = src4[lane * 2 + 1];
    unsafeAtomicAdd(dst + 0, v * p0.x);
    unsafeAtomicAdd(dst + 1, v * p0.y);
    unsafeAtomicAdd(dst + 2, v * p0.z);
    unsafeAtomicAdd(dst + 3, v * p0.w);
    unsafeAtomicAdd(dst + 4, v * p1.x);
    unsafeAtomicAdd(dst + 5, v * p1.y);
    unsafeAtomicAdd(dst + 6, v * p1.z);
    unsafeAtomicAdd(dst + 7, v * p1.w);
}

// ---------------- fused dual-GEMM via V_WMMA_F32_16X16X4_F32 ----
// out = (emb+nb) @ W1 + (emb*nb) @ W2
// block: 512 threads = 16 waves; one 16-row stripe; wave w owns cols [16w,16w+16)
__global__ __launch_bounds__(512) void ngcf_gemm_kernel(const float* __restrict__ emb,
                                                        const float* __restrict__ nb,
                                                        const float* __restrict__ W1,
                                                        const float* __restrict__ W2,
                                                        float* __restrict__ out, int N) {
    __shared__ float As[16 * 257];   // emb + nb   (row stride 257: conflict-free)
    __shared__ float Ap[16 * 257];   // emb * nb
    const int rowBase = blockIdx.x * 16;
    const int tid = threadIdx.x;

    // cooperative tile build: 4096 elements, 8 per thread
#pragma unroll
    for (int j = 0; j < 8; ++j) {
        int idx = tid + 512 * j;
        int r = idx >> 8;
        int cc = idx & 255;
        int gr = rowBase + r;
        if (gr >= N) gr = N - 1;   // tail clamp (N=100000 is divisible; cheap guard)
        size_t g = (size_t)gr * DIMD + cc;
        float e = emb[g];
        float n = nb[g];
        As[r * 257 + cc] = e + n;
        Ap[r * 257 + cc] = e * n;
    }
    __syncthreads();

    const int lane = tid & 31;
    const int wave = tid >> 5;          // 0..15 -> column tile
    const int colBase = wave * 16;
    const int m   = lane & 15;          // A row / B-C-D column within tile
    const int kh  = (lane >> 4) * 2;    // 0 or 2: K-pair held by this half-wave

    v8f c = {};
#pragma unroll 4
    for (int k = 0; k < DIMD; k += 4) {
        v2f a1, a2, b1, b2;
        a1.x = As[m * 257 + k + kh];
        a1.y = As[m * 257 + k + kh + 1];
        a2.x = Ap[m * 257 + k + kh];
        a2.y = Ap[m * 257 + k + kh + 1];
        b1.x = W1[(size_t)(k + kh) * DIMD + colBase + m];
        b1.y = W1[(size_t)(k + kh + 1) * DIMD + colBase + m];
        b2.x = W2[(size_t)(k + kh) * DIMD + colBase + m];
        b2.y = W2[(size_t)(k + kh + 1) * DIMD + colBase + m];
        // D = A*B + C, fp32 throughout (matches reference precision)
        c = __builtin_amdgcn_wmma_f32_16x16x4_f32(false, a1, false, b1, (short)0, c, false, false);
        c = __builtin_amdgcn_wmma_f32_16x16x4_f32(false, a2, false, b2, (short)0, c, false, false);
    }

    // C/D layout: VGPR r -> M = r + 8*(lane>=16), N = lane&15.
    // One base pointer + immediate offsets (r*1024B fits the 24-bit IOFFSET);
    // tail guard is uniform so the hot path is 8 plain global_store_b32.
    const int orow0 = rowBase + (lane >> 4) * 8;
    float* op = out + (size_t)orow0 * DIMD + colBase + m;
    if (rowBase + 16 <= N) {
#pragma unroll
        for (int r = 0; r < 8; ++r) op[(size_t)r * DIMD] = c[r];
    } else {
#pragma unroll
        for (int r = 0; r < 8; ++r)
            if (orow0 + r < N) op[(size_t)r * DIMD] = c[r];
    }
}

extern "C" void kernel_launch(void* const* d_in, const int* in_sizes, int n_in,
                              void* d_out, int out_size, void* d_ws, size_t ws_size,
                              hipStream_t stream) {
    const float* emb  = (const float*)d_in[0];
    const float* vals = (const float*)d_in[1];
    const float* W1   = (const float*)d_in[2];
    const float* W2   = (const float*)d_in[3];
    const int*   rows = (const int*)d_in[4];
    const int*   cols = (const int*)d_in[5];
    float* out = (float*)d_out;

    const int N = in_sizes[0] / DIMD;
    const int E = in_sizes[1];

    // neighbor accumulator: workspace if large enough, else d_out
    // (d_out path is safe: GEMM stages its 16-row tile to LDS + barrier
    //  before overwriting those same rows)
    const size_t need = (size_t)N * DIMD * sizeof(float);
    float* nb = (ws_size >= need) ? (float*)d_ws : out;

    // 1) nb = 0
    {
        int n4 = N * (DIMD / 4);
        zero_kernel<<<(n4 + 255) / 256, 256, 0, stream>>>((float4*)nb, n4);
    }
    // 2) nb = A @ emb   (COO scatter, 1 wave per edge)
    {
        int blocks = (E + 7) / 8;
        spmm_kernel<<<blocks, 256, 0, stream>>>(emb, vals, rows, cols, nb, E);
    }
    // 3) out = (emb+nb)@W1 + (emb*nb)@W2
    {
        int blocks = (N + 15) / 16;
        ngcf_gemm_kernel<<<blocks, 512, 0, stream>>>(emb, nb, W1, W2, out, N);
    }
}